// TripletLoss_60232621359473
// MI455X (gfx1250) — compile-verified
//
#include <hip/hip_runtime.h>

typedef __attribute__((ext_vector_type(2))) float v2f;
typedef __attribute__((ext_vector_type(8))) float v8f;

#define BN 512
#define DN 512
#define MARGIN_F 0.2f
#define INF_F 1e30f

// ---------------------------------------------------------------------------
// Wave32 reductions (CDNA5 is wave32; use width-32 shuffles explicitly).
// ---------------------------------------------------------------------------
__device__ __forceinline__ float waveReduceSum(float v) {
#pragma unroll
  for (int off = 16; off > 0; off >>= 1) v += __shfl_xor(v, off, 32);
  return v;
}

__device__ __forceinline__ float waveReduceMin(float v) {
#pragma unroll
  for (int off = 16; off > 0; off >>= 1) v = fminf(v, __shfl_xor(v, off, 32));
  return v;
}

// Block-wide min over blockDim.x (multiple of 32) threads; result broadcast.
// Safe to call repeatedly in a uniform loop (syncs bracket the shared slot).
__device__ __forceinline__ float blockReduceMin(float v, float* red, int nwaves) {
  v = waveReduceMin(v);
  const int wid = threadIdx.x >> 5;
  const int lid = threadIdx.x & 31;
  if (lid == 0) red[wid] = v;
  __syncthreads();
  if (wid == 0) {
    float x = (lid < nwaves) ? red[lid] : INF_F;
#pragma unroll
    for (int off = 4; off > 0; off >>= 1) x = fminf(x, __shfl_xor(x, off, 32));
    if (lid == 0) red[0] = x;
  }
  __syncthreads();
  float r = red[0];
  __syncthreads();
  return r;
}

// ---------------------------------------------------------------------------
// Kernel 0: zero the scalar accumulators (graph-safe, deterministic).
// ---------------------------------------------------------------------------
__global__ void tl_init_kernel(float* __restrict__ acc) {
  acc[0] = 0.0f;  // loss sum
  acc[1] = 0.0f;  // valid-pair count
}

// ---------------------------------------------------------------------------
// Kernel 1: per-row squared norms. One block (4 waves) per row.
// ---------------------------------------------------------------------------
__global__ void tl_sqnorm_kernel(const float* __restrict__ X,
                                 float* __restrict__ sq) {
  __shared__ float red[4];
  const int row = blockIdx.x;
  const int t = threadIdx.x;  // 128 threads
  float s = 0.0f;
  for (int d = t; d < DN; d += 128) {
    float v = X[row * DN + d];
    s = fmaf(v, v, s);
  }
  s = waveReduceSum(s);
  const int wid = t >> 5, lid = t & 31;
  if (lid == 0) red[wid] = s;
  __syncthreads();
  if (t == 0) sq[row] = red[0] + red[1] + red[2] + red[3];
}

// ---------------------------------------------------------------------------
// Kernel 2: Gram matrix via V_WMMA_F32_16X16X4_F32, fused into the squared
// distance transform. One wave per 16x16 tile of Dm. Because B[k][n] =
// X[J+n][k] for the Gram matrix, the A and B fragments use the identical
// per-lane load pattern (ISA 16x4 f32 A layout: lane l%16 = row, l/16 picks
// the K pair), so every lane issues one contiguous float2 load per fragment.
// ---------------------------------------------------------------------------
__global__ void tl_gram_dist_kernel(const float* __restrict__ X,
                                    const float* __restrict__ sq,
                                    float* __restrict__ Dm) {
  const int I = blockIdx.x * 16;
  const int J = blockIdx.y * 16;
  const int lane = threadIdx.x;      // 0..31, one wave
  const int r16 = lane & 15;
  const int hi = lane >> 4;          // 0: K pair {0,1}, 1: K pair {2,3}

  const float* arow = X + (size_t)(I + r16) * DN + 2 * hi;
  const float* brow = X + (size_t)(J + r16) * DN + 2 * hi;

  v8f c = {};
  for (int k = 0; k < DN; k += 4) {
    v2f a = *(const v2f*)(arow + k);
    v2f b = *(const v2f*)(brow + k);
    // D = A(16x4) * B(4x16) + C, full f32.
    c = __builtin_amdgcn_wmma_f32_16x16x4_f32(
        /*neg_a=*/false, a, /*neg_b=*/false, b,
        /*c_mod=*/(short)0, c, /*reuse_a=*/false, /*reuse_b=*/false);
  }

  // C/D layout: VGPR r -> M = r + 8*(lane/16), N = lane%16.
  const int n = lane & 15;
  const int mbase = hi * 8;
  const float sqn = sq[J + n];
#pragma unroll
  for (int rr = 0; rr < 8; ++rr) {
    const int m = mbase + rr;
    float d = sq[I + m] + sqn - 2.0f * c[rr];
    Dm[(size_t)(I + m) * BN + (J + n)] = fmaxf(d, 0.0f);
  }
}

// ---------------------------------------------------------------------------
// Kernel 3: semi-hard mining. One block (8 waves) per anchor row. Row and
// labels cached in LDS; label tests are uniform across the block, so the
// positive loop never diverges. min-value == Dm[neg_idx] (argmin over values
// equal to their own Dm entries), so we never materialize indices.
// ---------------------------------------------------------------------------
#define MT 256
__global__ void tl_mine_kernel(const float* __restrict__ Dm,
                               const int* __restrict__ labels,
                               float* __restrict__ acc) {
  __shared__ float row[BN];
  __shared__ int lab[BN];
  __shared__ float red[MT / 32];

  const int a = blockIdx.x;
  const int t = threadIdx.x;

  for (int n = t; n < BN; n += MT) {
    row[n] = Dm[(size_t)a * BN + n];
    lab[n] = labels[n];
  }
  __syncthreads();

  const int la = lab[a];

  // Hardest-overall negative of this row (fallback when no semi-hard exists).
  float m = INF_F;
  for (int n = t; n < BN; n += MT)
    if (lab[n] != la) m = fminf(m, row[n]);
  const float minNeg = blockReduceMin(m, red, MT / 32);
  const bool hasNeg = (minNeg < 0.5f * INF_F);

  float lsum = 0.0f, lcnt = 0.0f;
  for (int p = a + 1; p < BN; ++p) {
    if (lab[p] != la) continue;  // uniform: lab[] is in LDS
    const float ap = row[p];
    float mm = INF_F;
    for (int n = t; n < BN; n += MT) {
      const float dn = row[n];
      if (lab[n] != la && dn > ap && dn < ap + MARGIN_F) mm = fminf(mm, dn);
    }
    mm = blockReduceMin(mm, red, MT / 32);
    if (t == 0 && hasNeg) {
      const float negd = (mm < 0.5f * INF_F) ? mm : minNeg;
      lsum += fmaxf(ap - negd + MARGIN_F, 0.0f);
      lcnt += 1.0f;
    }
  }
  if (t == 0) {
    atomicAdd(&acc[0], lsum);
    atomicAdd(&acc[1], lcnt);
  }
}

// ---------------------------------------------------------------------------
// Kernel 4: finalize mean.
// ---------------------------------------------------------------------------
__global__ void tl_finalize_kernel(const float* __restrict__ acc,
                                   float* __restrict__ out) {
  out[0] = acc[0] / fmaxf(acc[1], 1.0f);
}

// ---------------------------------------------------------------------------
// Host entry. Workspace layout: Dm (512*512 f32) | sq (512 f32) | acc (2 f32).
// ---------------------------------------------------------------------------
extern "C" void kernel_launch(void* const* d_in, const int* in_sizes, int n_in,
                              void* d_out, int out_size, void* d_ws,
                              size_t ws_size, hipStream_t stream) {
  (void)in_sizes; (void)n_in; (void)out_size; (void)ws_size;
  const float* X = (const float*)d_in[0];
  const int* labels = (const int*)d_in[1];

  float* Dm = (float*)d_ws;
  float* sq = Dm + (size_t)BN * BN;
  float* acc = sq + BN;

  tl_init_kernel<<<1, 1, 0, stream>>>(acc);
  tl_sqnorm_kernel<<<BN, 128, 0, stream>>>(X, sq);
  dim3 grid(BN / 16, BN / 16);
  tl_gram_dist_kernel<<<grid, 32, 0, stream>>>(X, sq, Dm);
  tl_mine_kernel<<<BN, MT, 0, stream>>>(Dm, labels, acc);
  tl_finalize_kernel<<<1, 1, 0, stream>>>(acc, (float*)d_out);
}